// Yolov3Detector_86552180949073
// MI455X (gfx1250) — compile-verified
//
#include <hip/hip_runtime.h>

// CDNA5 / gfx1250: wave32, WMMA f32 16x16x4.
// A (16x4 f32): lane = M row (lane&15); VGPR0 = K + 2*(lane>>4), VGPR1 = that + 1.
// B (4x16 f32): mirrored K split across lane halves; N = lane&15 within each VGPR.
// C/D (16x16 f32): VGPR r -> M = r + 8*(lane>>4), N = lane&15.
//
// Shapes are template parameters so all strides are compile-time constants:
// B-operand loads become base + immediate-offset (<= 2.77 MB span, fits the
// signed 24-bit IOFFSET), removing per-iteration 64-bit address adds, and the
// epilogue's n/S and s/GRID divisions become constant magic-multiplies.

typedef __attribute__((ext_vector_type(2))) float v2f;
typedef __attribute__((ext_vector_type(8))) float v8f;

__device__ __forceinline__ float sigf(float x) {
    // 1 / (1 + e^-x) via the fast HW rcp/exp paths
    return __builtin_amdgcn_rcpf(1.0f + __expf(-x));
}

template <int C, int S, int G>
__global__ __launch_bounds__(256)
void yolo_head_wmma(const float* __restrict__ F,    // (32, C, S) activations
                    const float* __restrict__ Wm,   // (255, C) head weights
                    const float* __restrict__ bias, // (255)
                    float* __restrict__ out,        // (32, 3*S, 85) head output base
                    float aw0, float ah0, float aw1, float ah1,
                    float aw2, float ah2)
{
    constexpr float inv_grid = 1.0f / (float)G;

    const int lane  = threadIdx.x & 31;
    const int wave  = (blockIdx.x << 3) | (threadIdx.x >> 5);
    const int tm    = wave & 15;        // 16 M-tiles cover channels 0..254 (last clamped)
    const int tn    = wave >> 4;        // N-tile over flattened (batch, spatial)

    const int l16   = lane & 15;
    const int half  = lane >> 4;
    const int half2 = half << 1;        // K sub-offset: 0 (lanes 0-15) or 2 (lanes 16-31)

    const int o_base = tm << 4;
    const int o_row  = o_base + l16;
    const int o_load = o_row > 254 ? 254 : o_row;   // clamp reads; masked at store

    const int n = (tn << 4) + l16;      // flattened (b, s); 32*S is a multiple of 16
    const int b = n / S;                // constant divisor -> magic multiply
    const int s = n - b * S;

    const float* __restrict__ wp = Wm + (size_t)o_load * C + half2;
    const float* __restrict__ fp = F + (size_t)b * C * S + (size_t)half2 * S + s;

    v8f acc = {0.f, 0.f, 0.f, 0.f, 0.f, 0.f, 0.f, 0.f};

    #pragma unroll 8
    for (int k = 0; k < C; k += 4) {    // C/4 iterations, divisible by 8: no tail
        v2f a = *(const v2f*)(wp + k);  // W[o, k+half2 .. +1]  (8B aligned)
        v2f bb;
        bb.x = fp[k * S];               // F[b, k+half2,   s]  (immediate offsets)
        bb.y = fp[k * S + S];           // F[b, k+half2+1, s]
        acc = __builtin_amdgcn_wmma_f32_16x16x4_f32(
                  false, a, false, bb, (short)0, acc, false, false);
    }

    // ---- fused YOLO decode epilogue ----
    const int h = s / G;                // constant divisor
    const int w = s - h * G;
    const float wcol = (float)w;
    const float hrow = (float)h;
    float* __restrict__ obase = out + (size_t)b * 3 * S * 85 + (size_t)s * 85;

    const int mofs = half << 3;         // 0 or 8
    #pragma unroll
    for (int r = 0; r < 8; ++r) {
        const int o = o_base + mofs + r;    // output channel = a*85 + attr
        if (o < 255) {
            const float v = acc[r] + bias[o];
            const int   a    = o / 85;          // anchor index 0..2
            const int   attr = o - a * 85;      // 0..84
            const float aw = (a == 0) ? aw0 : ((a == 1) ? aw1 : aw2);
            const float ah = (a == 0) ? ah0 : ((a == 1) ? ah1 : ah2);
            float res;
            if (attr >= 4)      res = sigf(v);                     // conf + 80 classes
            else if (attr == 0) res = (wcol + sigf(v)) * inv_grid; // bx
            else if (attr == 1) res = (hrow + sigf(v)) * inv_grid; // by
            else if (attr == 2) res = __expf(v) * aw;              // bw (anchor/416 baked)
            else                res = __expf(v) * ah;              // bh
            // Write-once streaming output: non-temporal, keep L2 for A/B reuse
            __builtin_nontemporal_store(res, obase + (size_t)a * S * 85 + attr);
        }
    }
}

template <int C, int S, int G>
static void launch_head(const float* F, const float* Wm, const float* bias,
                        float* out, const float* anch6, hipStream_t stream)
{
    constexpr int N      = 32 * S;       // flattened (batch, spatial); multiple of 16
    constexpr int waves  = 16 * (N / 16);// 16 M-tiles x N/16 N-tiles
    constexpr int blocks = waves / 8;    // 8 waves (256 threads) per block, exact
    const float k = 1.0f / 416.0f;
    yolo_head_wmma<C, S, G><<<dim3(blocks), dim3(256), 0, stream>>>(
        F, Wm, bias, out,
        anch6[0] * k, anch6[1] * k, anch6[2] * k,
        anch6[3] * k, anch6[4] * k, anch6[5] * k);
}

extern "C" void kernel_launch(void* const* d_in, const int* in_sizes, int n_in,
                              void* d_out, int out_size, void* d_ws, size_t ws_size,
                              hipStream_t stream) {
    (void)in_sizes; (void)n_in; (void)out_size; (void)d_ws; (void)ws_size;
    const float* f13 = (const float*)d_in[0];
    const float* f26 = (const float*)d_in[1];
    const float* f52 = (const float*)d_in[2];
    const float* W13 = (const float*)d_in[3];
    const float* b13 = (const float*)d_in[4];
    const float* W26 = (const float*)d_in[5];
    const float* b26 = (const float*)d_in[6];
    const float* W52 = (const float*)d_in[7];
    const float* b52 = (const float*)d_in[8];
    float* out = (float*)d_out;

    // stride-32 head: anchors rows [6,7,8]; stride-16: [3,4,5]; stride-8: [0,1,2]
    static const float A13[6] = {116.f, 90.f, 156.f, 198.f, 373.f, 326.f};
    static const float A26[6] = { 30.f, 61.f,  62.f,  45.f,  59.f, 119.f};
    static const float A52[6] = { 10.f, 13.f,  16.f,  30.f,  33.f,  23.f};

    // Output tuple concatenated flat: out13 | out26 | out52
    const size_t off13 = 0;
    const size_t off26 = (size_t)32 * 507  * 85;          //  1,379,040
    const size_t off52 = off26 + (size_t)32 * 2028 * 85;  // + 5,516,160

    launch_head<1024, 13 * 13, 13>(f13, W13, b13, out + off13, A13, stream);
    launch_head< 512, 26 * 26, 26>(f26, W26, b26, out + off26, A26, stream);
    launch_head< 256, 52 * 52, 52>(f52, W52, b52, out + off52, A52, stream);
}